// GraphConv_13048110645351
// MI455X (gfx1250) — compile-verified
//
#include <hip/hip_runtime.h>
#include <hip/hip_bf16.h>
#include <math.h>

#define N_USERS    50000
#define N_ENTITIES 100000
#define CHANNEL    64
#define N_FACTORS  4
#define N_RELM1    8
#define N_EDGES    1500000
#define NNZ        800000
#define N_HOPS     2

typedef __attribute__((ext_vector_type(2))) float v2f;
typedef __attribute__((ext_vector_type(8))) float v8f;

// Native fp32 atomic add (global_atomic_add_f32) — destinations fit in L2.
__device__ __forceinline__ void atomAddF(float* p, float v) {
    unsafeAtomicAdd(p, v);
}

// ---------------------------------------------------------------- utilities
// float4 streaming copy: src -> d1 and d2 (element counts divisible by 4)
__global__ void k_copy2_v4(const float4* __restrict__ src, float4* __restrict__ d1,
                           float4* __restrict__ d2, int n4) {
    int i = blockIdx.x * blockDim.x + threadIdx.x;
    if (i < n4) { float4 v = src[i]; d1[i] = v; d2[i] = v; }
}

__global__ void k_zero_v4(float4* __restrict__ p, int n4) {
    int i = blockIdx.x * blockDim.x + threadIdx.x;
    if (i < n4) p[i] = make_float4(0.f, 0.f, 0.f, 0.f);
}

__global__ void k_count(const int* __restrict__ head, float* __restrict__ cnt, int n) {
    int i = blockIdx.x * blockDim.x + threadIdx.x;
    if (i < n) atomAddF(&cnt[head[i]], 1.0f);
}

// ---------------------------------------------------------------- cor (tiny)
__global__ void k_cor(const float* __restrict__ att, float* __restrict__ out) {
    if (threadIdx.x != 0) return;
    const int ch = N_RELM1; // 8
    float cor = 0.0f;
    for (int i = 0; i < N_FACTORS; ++i) {
        for (int j = i + 1; j < N_FACTORS; ++j) {
            const float* t1 = att + i * ch;
            const float* t2 = att + j * ch;
            float a[8][8], b[8][8];
            for (int m = 0; m < ch; ++m)
                for (int n = 0; n < ch; ++n) {
                    float d1 = t1[m]*t1[m] - 2.0f*t1[m]*t1[n] + t1[n]*t1[n];
                    float d2 = t2[m]*t2[m] - 2.0f*t2[m]*t2[n] + t2[n]*t2[n];
                    a[m][n] = sqrtf(fmaxf(d1, 0.0f) + 1e-8f);
                    b[m][n] = sqrtf(fmaxf(d2, 0.0f) + 1e-8f);
                }
            float am0[8] = {0}, am1[8] = {0}, bm0[8] = {0}, bm1[8] = {0};
            float ga = 0.0f, gb = 0.0f;
            for (int m = 0; m < ch; ++m)
                for (int n = 0; n < ch; ++n) {
                    am0[n] += a[m][n]; am1[m] += a[m][n]; ga += a[m][n];
                    bm0[n] += b[m][n]; bm1[m] += b[m][n]; gb += b[m][n];
                }
            ga /= (float)(ch * ch); gb /= (float)(ch * ch);
            float sAB = 0, sAA = 0, sBB = 0;
            for (int m = 0; m < ch; ++m)
                for (int n = 0; n < ch; ++n) {
                    float A = a[m][n] - am0[n]/(float)ch - am1[m]/(float)ch + ga;
                    float B = b[m][n] - bm0[n]/(float)ch - bm1[m]/(float)ch + gb;
                    sAB += A*B; sAA += A*A; sBB += B*B;
                }
            float dAB = sqrtf(fmaxf(sAB / (float)(ch*ch), 0.0f) + 1e-8f);
            float dAA = sqrtf(fmaxf(sAA / (float)(ch*ch), 0.0f) + 1e-8f);
            float dBB = sqrtf(fmaxf(sBB / (float)(ch*ch), 0.0f) + 1e-8f);
            cor += dAB / sqrtf(dAA * dBB + 1e-8f);
        }
    }
    out[0] = cor;
}

// -------------------------------------------- disen_weight = softmax(att)@W (4x64)
__global__ void k_disen(const float* __restrict__ att, const float* __restrict__ weight,
                        float* __restrict__ dw) {
    int t = threadIdx.x;          // 256 threads
    int f = t >> 6, c = t & 63;
    const float* row = att + f * N_RELM1;
    float m = row[0];
    #pragma unroll
    for (int r = 1; r < N_RELM1; ++r) m = fmaxf(m, row[r]);
    float p[N_RELM1], s = 0.0f;
    #pragma unroll
    for (int r = 0; r < N_RELM1; ++r) { p[r] = expf(row[r] - m); s += p[r]; }
    float inv = 1.0f / s, acc = 0.0f;
    #pragma unroll
    for (int r = 0; r < N_RELM1; ++r) acc += p[r] * inv * weight[r * CHANNEL + c];
    dw[f * CHANNEL + c] = acc;
}

// ------------------------------------------------ edge scatter: wave per edge
__global__ void __launch_bounds__(256)
k_edge_scatter(const float* __restrict__ ent, const float* __restrict__ weight,
               const int* __restrict__ head, const int* __restrict__ tail,
               const int* __restrict__ etype, float* __restrict__ agg, int nE) {
    int e    = (blockIdx.x * blockDim.x + threadIdx.x) >> 5;
    int lane = threadIdx.x & 31;
    if (e >= nE) return;
    int h = head[e], t = tail[e], r = etype[e] - 1;
    float2 x = *(const float2*)(ent    + (size_t)t * CHANNEL + lane * 2);
    float2 w = *(const float2*)(weight + (size_t)r * CHANNEL + lane * 2);
    float* dst = agg + (size_t)h * CHANNEL + lane * 2;
    atomAddF(dst + 0, x.x * w.x);
    atomAddF(dst + 1, x.y * w.y);
}

// ------------------------------------------ interact scatter: wave per nnz
__global__ void __launch_bounds__(256)
k_interact_scatter(const float* __restrict__ ent, const float* __restrict__ vals,
                   const int* __restrict__ rows, const int* __restrict__ cols,
                   float* __restrict__ uagg, int n) {
    int e    = (blockIdx.x * blockDim.x + threadIdx.x) >> 5;
    int lane = threadIdx.x & 31;
    if (e >= n) return;
    int r = rows[e], c = cols[e];
    float v = vals[e];
    float2 x = *(const float2*)(ent + (size_t)c * CHANNEL + lane * 2);
    float* dst = uagg + (size_t)r * CHANNEL + lane * 2;
    atomAddF(dst + 0, v * x.x);
    atomAddF(dst + 1, v * x.y);
}

// ------------------------------------ entity update: wave per row, /cnt + l2norm
__global__ void __launch_bounds__(256)
k_entity_update(const float* __restrict__ agg, const float* __restrict__ cnt,
                float* __restrict__ cur, float* __restrict__ res, int nRows) {
    int row  = (blockIdx.x * blockDim.x + threadIdx.x) >> 5;
    int lane = threadIdx.x & 31;
    if (row >= nRows) return;
    float c = fmaxf(cnt[row], 1.0f);
    float2 v = *(const float2*)(agg + (size_t)row * CHANNEL + lane * 2);
    v.x /= c; v.y /= c;
    float ss = v.x * v.x + v.y * v.y;
    #pragma unroll
    for (int off = 16; off >= 1; off >>= 1) ss += __shfl_xor(ss, off, 32);
    float denom = fmaxf(sqrtf(ss), 1e-12f);
    float2 o; o.x = v.x / denom; o.y = v.y / denom;
    *(float2*)(cur + (size_t)row * CHANNEL + lane * 2) = o;
    float* rp = res + (size_t)row * CHANNEL + lane * 2;
    rp[0] += o.x; rp[1] += o.y;
}

// --------------- user update: softmax(U@L^T) then WMMA (16x16x4 f32) mix, norm
// Each wave handles a 16-user tile; mix = score(16x4) @ dw_tile(4x16) via
// V_WMMA_F32_16X16X4_F32 (K=4 exactly, fp32 end-to-end).
__global__ void __launch_bounds__(256)
k_user_update(const float* __restrict__ userIn, const float* __restrict__ latent,
              const float* __restrict__ dw, const float* __restrict__ uagg,
              float* __restrict__ userOut, float* __restrict__ userRes) {
    __shared__ float lds[8][16][CHANNEL + 1];   // +1 pad: avoid bank conflicts
    int wave = threadIdx.x >> 5;
    int lane = threadIdx.x & 31;
    int uBase = blockIdx.x * 128 + wave * 16;

    // Phase 1: lanes 0..15 each compute one user's 4-factor softmax score.
    float s0 = 0, s1 = 0, s2 = 0, s3 = 0;
    int myU = uBase + (lane & 15);
    if (lane < 16 && myU < N_USERS) {
        const float* ur = userIn + (size_t)myU * CHANNEL;
        float d0 = 0, d1 = 0, d2 = 0, d3 = 0;
        for (int c = 0; c < CHANNEL; ++c) {
            float x = ur[c];
            d0 += x * latent[0 * CHANNEL + c];
            d1 += x * latent[1 * CHANNEL + c];
            d2 += x * latent[2 * CHANNEL + c];
            d3 += x * latent[3 * CHANNEL + c];
        }
        float m  = fmaxf(fmaxf(d0, d1), fmaxf(d2, d3));
        float e0 = expf(d0 - m), e1 = expf(d1 - m), e2 = expf(d2 - m), e3 = expf(d3 - m);
        float inv = 1.0f / (e0 + e1 + e2 + e3);
        s0 = e0 * inv; s1 = e1 * inv; s2 = e2 * inv; s3 = e3 * inv;
    }
    // A operand (16x4 f32): lanes 0-15 hold K=0/K=1, lanes 16-31 hold K=2/K=3.
    float s2o = __shfl(s2, lane & 15, 32);
    float s3o = __shfl(s3, lane & 15, 32);
    v2f A;
    A.x = (lane < 16) ? s0 : s2o;
    A.y = (lane < 16) ? s1 : s3o;

    // Phase 2: four 16-column tiles, one WMMA each.
    #pragma unroll
    for (int t = 0; t < 4; ++t) {
        int col = t * 16 + (lane & 15);
        // B operand (4x16 f32): lanes 0-15 K=0/K=1, lanes 16-31 K=2/K=3.
        v2f Bop;
        if (lane < 16) { Bop.x = dw[0 * CHANNEL + col]; Bop.y = dw[1 * CHANNEL + col]; }
        else           { Bop.x = dw[2 * CHANNEL + col]; Bop.y = dw[3 * CHANNEL + col]; }
        v8f C = {};
        C = __builtin_amdgcn_wmma_f32_16x16x4_f32(false, A, false, Bop,
                                                  (short)0, C, false, false);
        // D layout: VGPR r, lanes 0-15 -> row r; lanes 16-31 -> row r+8.
        // Clamp row index so the uagg load is unconditional (no exec-mask
        // branches); garbage rows only reach LDS cells whose final stores
        // are guarded in phase 3.
        #pragma unroll
        for (int r = 0; r < 8; ++r) {
            int row = (lane < 16) ? r : (r + 8);
            int uu  = uBase + row;
            int uc  = (uu < N_USERS) ? uu : (N_USERS - 1);
            float mix = C[r];
            float ua  = uagg[(size_t)uc * CHANNEL + col];
            lds[wave][row][col] = ua * mix + ua;   // user_agg*mix + user_agg
        }
    }
    __syncthreads();

    // Phase 3: l2-normalize rows (2 lanes per row), write emb + residual.
    int row  = lane >> 1;
    int half = lane & 1;
    const float* lr = &lds[wave][row][half * 32];
    float ss = 0.0f;
    #pragma unroll
    for (int c = 0; c < 32; ++c) { float x = lr[c]; ss += x * x; }
    ss += __shfl_xor(ss, 1, 32);
    float denom = fmaxf(sqrtf(ss), 1e-12f);
    int uo = uBase + row;
    if (uo < N_USERS) {
        float* co = userOut + (size_t)uo * CHANNEL + half * 32;
        float* ro = userRes + (size_t)uo * CHANNEL + half * 32;
        #pragma unroll
        for (int c = 0; c < 32; ++c) {
            float v = lr[c] / denom;
            co[c] = v;
            ro[c] += v;
        }
    }
}

// ---------------------------------------------------------------- launcher
extern "C" void kernel_launch(void* const* d_in, const int* in_sizes, int n_in,
                              void* d_out, int out_size, void* d_ws, size_t ws_size,
                              hipStream_t stream) {
    const float* user_emb   = (const float*)d_in[0];
    const float* entity_emb = (const float*)d_in[1];
    const float* latent     = (const float*)d_in[2];
    const float* weight     = (const float*)d_in[3];
    const float* att        = (const float*)d_in[4];
    const float* ivals      = (const float*)d_in[5];
    const int*   head       = (const int*)d_in[6];
    const int*   tail       = (const int*)d_in[7];
    const int*   etype      = (const int*)d_in[8];
    const int*   irow       = (const int*)d_in[9];
    const int*   icol       = (const int*)d_in[10];

    float* out    = (float*)d_out;
    float* entRes = out;
    float* usrRes = out + (size_t)N_ENTITIES * CHANNEL;
    float* corOut = usrRes + (size_t)N_USERS * CHANNEL;

    float* ws = (float*)d_ws;
    size_t off = 0;
    float* entCur = ws + off; off += (size_t)N_ENTITIES * CHANNEL;
    float* usrCur = ws + off; off += (size_t)N_USERS * CHANNEL;
    float* entAgg = ws + off; off += (size_t)N_ENTITIES * CHANNEL;   // contiguous
    float* usrAgg = ws + off; off += (size_t)N_USERS * CHANNEL;      // with entAgg
    float* cnt    = ws + off; off += N_ENTITIES;
    float* dw     = ws + off; off += N_FACTORS * CHANNEL;

    const int B = 256;
    const int nE64 = N_ENTITIES * CHANNEL;     // 6.4M
    const int nU64 = N_USERS * CHANNEL;        // 3.2M
    const int nAgg = nE64 + nU64;              // contiguous agg region

    k_copy2_v4<<<(nE64/4 + B - 1) / B, B, 0, stream>>>(
        (const float4*)entity_emb, (float4*)entCur, (float4*)entRes, nE64 / 4);
    k_copy2_v4<<<(nU64/4 + B - 1) / B, B, 0, stream>>>(
        (const float4*)user_emb, (float4*)usrCur, (float4*)usrRes, nU64 / 4);
    k_zero_v4<<<(N_ENTITIES/4 + B - 1) / B, B, 0, stream>>>((float4*)cnt, N_ENTITIES / 4);
    k_count<<<(N_EDGES + B - 1) / B, B, 0, stream>>>(head, cnt, N_EDGES);
    k_cor  <<<1, 32, 0, stream>>>(att, corOut);
    k_disen<<<1, 256, 0, stream>>>(att, weight, dw);

    for (int hop = 0; hop < N_HOPS; ++hop) {
        k_zero_v4<<<(nAgg/4 + B - 1) / B, B, 0, stream>>>((float4*)entAgg, nAgg / 4);
        k_edge_scatter<<<(N_EDGES * 32 + B - 1) / B, B, 0, stream>>>(
            entCur, weight, head, tail, etype, entAgg, N_EDGES);
        k_interact_scatter<<<(NNZ * 32 + B - 1) / B, B, 0, stream>>>(
            entCur, ivals, irow, icol, usrAgg, NNZ);
        k_user_update<<<(N_USERS + 127) / 128, 256, 0, stream>>>(
            usrCur, latent, dw, usrAgg, usrCur, usrRes);
        k_entity_update<<<(N_ENTITIES * 32 + B - 1) / B, B, 0, stream>>>(
            entAgg, cnt, entCur, entRes, N_ENTITIES);
    }
}